// gdfn_batch_mix_73194832658613
// MI455X (gfx1250) — compile-verified
//
#include <hip/hip_runtime.h>
#include <hip/hip_bf16.h>
#include <stdint.h>

// Problem constants (from reference setup_inputs)
#define NB   8
#define NC   3
#define NH   512
#define NP   514          // padded plane extent (H + 2*OFF)
#define TILE 16           // output tile
#define PR   22           // img_p halo tile extent in LDS
#define G1   20           // w3 / img_r grid extent
#define G2   18           // w_mod grid extent

typedef unsigned int v4u __attribute__((ext_vector_type(4)));
typedef int          v8i __attribute__((ext_vector_type(8)));
typedef int          v4i __attribute__((ext_vector_type(4)));

#if __has_builtin(__builtin_amdgcn_tensor_load_to_lds)
#define USE_TDM 1
#else
#define USE_TDM 0
#endif

__global__ __launch_bounds__(256)
void gdfn_fused_kernel(const float* __restrict__ img,
                       const float* __restrict__ noise,
                       const int*   __restrict__ ids_sel,
                       const int*   __restrict__ ids_dil,
                       float*       __restrict__ out)
{
    __shared__ float sp[NC][PR][PR];     // img_p halo tile            (5808 B)
    __shared__ float w3s[G1][G1][9];     // channel-min weights        (14400 B)
    __shared__ float imrs[G1][G1];       // img_r tile (0 = zero-pad)  (1600 B)
    __shared__ float wmods[G2][G2];      // mixed weights (0 = pad)    (1296 B)

    const int tid = threadIdx.x;
    const int oy0 = blockIdx.y * TILE;
    const int ox0 = blockIdx.x * TILE;
    const int b   = blockIdx.z;

    // LDS row r holds padded row p = prow0 + r (clamped tile origin so the
    // TDM never forms a negative global offset; bottom/right OOB reads
    // zero-fill per the TDM spec and land in unused LDS rows).
    const int prow0 = (oy0 - 2 < 0) ? 0 : (oy0 - 2);
    const int qcol0 = (ox0 - 2 < 0) ? 0 : (ox0 - 2);

    // ---- Phase 0a: DMA the 3x22x22 noise halo tile into LDS via the TDM ----
#if USE_TDM
    if (tid == 0) {
        unsigned lds_off = (unsigned)(uintptr_t)(&sp[0][0][0]);
        unsigned long long ga = (unsigned long long)(uintptr_t)noise +
            4ull * ((unsigned long long)b * (unsigned long long)(NC * NP * NP) +
                    (unsigned long long)prow0 * (unsigned long long)NP +
                    (unsigned long long)qcol0);
        // D# group 0: count=1, lds_addr, 57b global_addr, type=2 (bits 127:126)
        v4u g0 = { 1u, lds_off, (unsigned)ga,
                   (unsigned)(ga >> 32) | 0x80000000u };
        const unsigned d0s = (unsigned)NP;            // tensor_dim0_stride
        const unsigned d1s = (unsigned)(NP * NP);     // tensor_dim1_stride
        v8i g1;
        g1[0] = (int)(2u << 16);                              // data_size = 4B
        g1[1] = (int)(((unsigned)NP & 0xffffu) << 16);        // tensor_dim0 lo16
        g1[2] = (int)((((unsigned)NP >> 16) & 0xffffu) |
                      (((unsigned)NP & 0xffffu) << 16));      // dim0 hi / dim1 lo
        g1[3] = (int)((((unsigned)NP >> 16) & 0xffffu) |
                      ((unsigned)PR << 16));                  // dim1 hi / tile_dim0
        g1[4] = (int)((unsigned)PR | ((unsigned)NC << 16));   // tile_dim1 / tile_dim2
        g1[5] = (int)d0s;                                     // dim0_stride[31:0]
        g1[6] = (int)((d1s & 0xffffu) << 16);                 // dim0_str hi / dim1_str lo16
        g1[7] = (int)(d1s >> 16);                             // dim1_stride[47:16]
        v4i g2 = { NC, 0, 0, 0 };                             // tensor_dim2 = 3
        v4i g3 = { 0, 0, 0, 0 };
#if __clang_major__ >= 23
        v8i gz = { 0, 0, 0, 0, 0, 0, 0, 0 };
        __builtin_amdgcn_tensor_load_to_lds(g0, g1, g2, g3, gz, 0);
#else
        __builtin_amdgcn_tensor_load_to_lds(g0, g1, g2, g3, 0);
#endif
        __builtin_amdgcn_s_wait_tensorcnt(0);
    }
#endif
    // Prefetch the img halo rows while the TDM is in flight.
    if (tid < NC * PR) {
        int c = tid / PR, r = tid % PR;
        int p = prow0 + r;
        if (p >= 1 && p <= NH) {
            int qm = (qcol0 > 1) ? (qcol0 - 1) : 0;
            const float* pf = img + (((size_t)b * NC + c) * NH + (p - 1)) * NH + qm;
            __builtin_prefetch(pf, 0, 1);
        }
    }
    __syncthreads();

    // ---- Phase 0b: add img (interior of padded plane) into the LDS tile ----
    for (int idx = tid; idx < NC * PR * PR; idx += 256) {
        int c = idx / (PR * PR);
        int rem = idx % (PR * PR);
        int r = rem / PR, t = rem % PR;
        int p = prow0 + r, q = qcol0 + t;
#if USE_TDM
        if (p >= 1 && p <= NH && q >= 1 && q <= NH)
            sp[c][r][t] += img[(((size_t)b * NC + c) * NH + (p - 1)) * NH + (q - 1)];
#else
        float nv = (p < NP && q < NP)
                 ? noise[(((size_t)b * NC + c) * NP + p) * NP + q] : 0.0f;
        if (p >= 1 && p <= NH && q >= 1 && q <= NH)
            nv += img[(((size_t)b * NC + c) * NH + (p - 1)) * NH + (q - 1)];
        sp[c][r][t] = nv;
#endif
    }
    __syncthreads();

    // ---- Phase 1: w3 (channel-min Gaussian weights) and img_r on 20x20 ----
    for (int pos = tid; pos < G1 * G1; pos += 256) {
        int i = pos / G1, j = pos % G1;
        int y = oy0 - 2 + i, x = ox0 - 2 + j;
        float ir = 0.0f;                       // zero = img_r zero-padding
        if ((unsigned)y < (unsigned)NH && (unsigned)x < (unsigned)NH) {
            int ry = y - prow0, rx = x - qcol0;
            float v[NC][9], w3l[9];
            #pragma unroll
            for (int c = 0; c < NC; ++c) {
                float s1 = 0.0f, s2 = 0.0f;
                #pragma unroll
                for (int dy = 0; dy < 3; ++dy)
                    #pragma unroll
                    for (int dx = 0; dx < 3; ++dx) {
                        float t = sp[c][ry + dy][rx + dx];
                        v[c][dy * 3 + dx] = t;
                        s1 += t; s2 += t * t;
                    }
                float mean  = s1 * (1.0f / 9.0f);
                float var   = (s2 - 9.0f * mean * mean) * (1.0f / 8.0f); // ddof=1
                float i2v   = 0.5f / var;
                #pragma unroll
                for (int k = 0; k < 9; ++k) {
                    float d  = v[c][k] - mean;
                    float wv = __expf(-d * d * i2v);
                    w3l[k] = (c == 0) ? wv : fminf(w3l[k], wv);
                }
            }
            float num = 0.0f, den = 0.0f;
            #pragma unroll
            for (int k = 0; k < 9; ++k) {
                den += w3l[k];
                num += w3l[k] * (v[0][k] + v[1][k] + v[2][k]);
                w3s[i][j][k] = w3l[k];
            }
            ir = num / den;
        }
        imrs[i][j] = ir;
    }
    __syncthreads();

    // ---- Phase 2: argmin/argmax of img_r windows -> w_mod on 18x18 ----
    const bool dil = (ids_dil[b] != 0);
    for (int pos = tid; pos < G2 * G2; pos += 256) {
        int i = pos / G2, j = pos % G2;
        int y = oy0 - 1 + i, x = ox0 - 1 + j;
        float wm = 0.0f;                       // zero = w_mod zero-padding
        if ((unsigned)y < (unsigned)NH && (unsigned)x < (unsigned)NH) {
            float mn = 3.4e38f, mx = -3.4e38f;
            int kmin = 0, kmax = 0;
            #pragma unroll
            for (int dy = 0; dy < 3; ++dy)
                #pragma unroll
                for (int dx = 0; dx < 3; ++dx) {
                    int k = dy * 3 + dx;
                    float t = imrs[i + dy][j + dx];
                    if (t < mn) { mn = t; kmin = k; }   // first-occurrence argmin
                    if (t > mx) { mx = t; kmax = k; }   // first-occurrence argmax
                }
            int id = dil ? kmax : kmin;
            wm = w3s[i + 1][j + 1][id];
        }
        wmods[i][j] = wm;
    }
    __syncthreads();

    // ---- Phase 3: second weighted average on 16x16, select, uint8 cast ----
    const bool sel = (ids_sel[b] != 0);
    {
        int i = tid / TILE, j = tid % TILE;
        int y = oy0 + i, x = ox0 + j;
        int ry = y - prow0, rx = x - qcol0;
        float wk[9], den = 0.0f;
        #pragma unroll
        for (int dy = 0; dy < 3; ++dy)
            #pragma unroll
            for (int dx = 0; dx < 3; ++dx) {
                float t = wmods[i + dy][j + dx];
                wk[dy * 3 + dx] = t; den += t;
            }
        float invden = 1.0f / den;
        #pragma unroll
        for (int c = 0; c < NC; ++c) {
            float num = 0.0f;
            #pragma unroll
            for (int dy = 0; dy < 3; ++dy)
                #pragma unroll
                for (int dx = 0; dx < 3; ++dx)
                    num += wk[dy * 3 + dx] * sp[c][ry + dy][rx + dx];
            float val = sel ? (num * invden)
                            : img[(((size_t)b * NC + c) * NH + y) * NH + x];
            int q = (int)val;                   // uint8 cast semantics
            q = q < 0 ? 0 : (q > 255 ? 255 : q);
            out[(((size_t)b * NC + c) * NH + y) * NH + x] = (float)q;
        }
    }
}

extern "C" void kernel_launch(void* const* d_in, const int* in_sizes, int n_in,
                              void* d_out, int out_size, void* d_ws, size_t ws_size,
                              hipStream_t stream) {
    (void)in_sizes; (void)n_in; (void)d_ws; (void)ws_size; (void)out_size;
    const float* img   = (const float*)d_in[0];
    const float* noise = (const float*)d_in[1];
    const int*   sel   = (const int*)d_in[2];
    const int*   dil   = (const int*)d_in[3];
    float*       out   = (float*)d_out;

    dim3 grid(NH / TILE, NH / TILE, NB);   // 32 x 32 x 8 workgroups
    dim3 block(256);                       // 8 wave32 waves
    gdfn_fused_kernel<<<grid, block, 0, stream>>>(img, noise, sel, dil, out);
}